// DSCONV_31078383354703
// MI455X (gfx1250) — compile-verified
//
#include <hip/hip_runtime.h>

typedef __attribute__((ext_vector_type(2))) float v2f;
typedef __attribute__((ext_vector_type(4))) float v4f;
typedef __attribute__((ext_vector_type(8))) float v8f;

#define KT   9
#define PADH 4
#define CC   16
#define OC   16
#define HD   80
#define WD   80
#define LD   80
#define SP   (HD * WD * LD)      /* 512000: one channel plane / #positions */
#define TL   (LD / 16)           /* 5 tiles along L */
#define NT   (HD * WD * TL)      /* 32000 tiles total */

// ---------------------------------------------------------------------------
// Pre-pass: x[C][H][W][L] -> xt[H][W][L][C]  (channel-last, 64B per position)
// Reads coalesced per channel plane, writes 4x b128 contiguous per lane.
// ---------------------------------------------------------------------------
__global__ __launch_bounds__(256)
void pack_xt(const float* __restrict__ x, float* __restrict__ xt) {
    const int pos = blockIdx.x * blockDim.x + threadIdx.x;
    if (pos >= SP) return;
    float v[CC];
    #pragma unroll
    for (int c = 0; c < CC; ++c) v[c] = x[(size_t)c * SP + pos];
    float* dst = xt + (size_t)pos * CC;
    #pragma unroll
    for (int c = 0; c < CC; c += 4) {
        v4f t = { v[c], v[c + 1], v[c + 2], v[c + 3] };
        *(v4f*)(dst + c) = t;
    }
}

// ---------------------------------------------------------------------------
// Main kernel, packed layout. Channel permutation for WMMA K-chunks:
//   chunk q covers channels {2q, 2q+1, 8+2q, 8+2q+1}
// so a lane in half h owns channels 8h..8h+7 (contiguous in xt) and each
// bilinear corner is two 16B vector loads.
// ---------------------------------------------------------------------------
__global__ __launch_bounds__(256)
void dsconv_wmma_packed(const float* __restrict__ xt,
                        const float* __restrict__ off,
                        const float* __restrict__ wgt,
                        float* __restrict__ out) {
    const int lane  = threadIdx.x & 31;
    const int half  = lane >> 4;
    const int p     = lane & 15;
    const int wpb   = blockDim.x >> 5;
    const int gwave = blockIdx.x * wpb + (threadIdx.x >> 5);
    const int nwav  = gridDim.x * wpb;
    const int cbase = 8 * half;              // lane's first channel in xt

    // A fragments with the permuted channel mapping:
    // afrag[k][q] = { W[o][8h+2q][k], W[o][8h+2q+1][k] }
    v2f afrag[KT][4];
    {
        const int o = p;
        #pragma unroll
        for (int k = 0; k < KT; ++k) {
            #pragma unroll
            for (int q = 0; q < 4; ++q) {
                const int c0 = cbase + 2 * q;
                afrag[k][q].x = wgt[(o * CC + c0    ) * KT + k];
                afrag[k][q].y = wgt[(o * CC + c0 + 1) * KT + k];
            }
        }
    }

    for (int t = gwave; t < NT; t += nwav) {
        const int l0 = (t % TL) * 16;
        const int w  = (t / TL) % WD;
        const int h  =  t / (TL * WD);
        const int l  = l0 + p;
        const int sbase = h * (WD * LD) + w * LD + l;
        const int huni  = __builtin_amdgcn_readfirstlane(h);

        v8f acc = {};

        #pragma unroll
        for (int k = 0; k < KT; ++k) {
            const int hk = huni + k - PADH;       // scalar-uniform tap validity
            if (hk < 0 || hk >= HD) continue;

            const float ow = off[(2 * k + 0) * SP + sbase];
            const float ol = off[(2 * k + 1) * SP + sbase];
            const float pw = (float)w + ow;
            const float pl = (float)l + ol;
            const float fw = floorf(pw);
            const float fl = floorf(pl);
            const float aw = pw - fw;
            const float al = pl - fl;
            const int iw0 = (int)fw, iw1 = iw0 + 1;
            const int il0 = (int)fl, il1 = il0 + 1;
            const bool vw0 = (iw0 >= 0) & (iw0 < WD);
            const bool vw1 = (iw1 >= 0) & (iw1 < WD);
            const bool vl0 = (il0 >= 0) & (il0 < LD);
            const bool vl1 = (il1 >= 0) & (il1 < LD);
            float wt00 = (1.f - aw) * (1.f - al); if (!(vw0 & vl0)) wt00 = 0.f;
            float wt01 = (1.f - aw) * al;         if (!(vw0 & vl1)) wt01 = 0.f;
            float wt10 = aw * (1.f - al);         if (!(vw1 & vl0)) wt10 = 0.f;
            float wt11 = aw * al;                 if (!(vw1 & vl1)) wt11 = 0.f;
            const int cw0 = min(max(iw0, 0), WD - 1);
            const int cw1 = min(max(iw1, 0), WD - 1);
            const int cl0 = min(max(il0, 0), LD - 1);
            const int cl1 = min(max(il1, 0), LD - 1);
            const int hb  = hk * (WD * LD);
            const size_t b00 = (size_t)(hb + cw0 * LD + cl0) * CC + cbase;
            const size_t b01 = (size_t)(hb + cw0 * LD + cl1) * CC + cbase;
            const size_t b10 = (size_t)(hb + cw1 * LD + cl0) * CC + cbase;
            const size_t b11 = (size_t)(hb + cw1 * LD + cl1) * CC + cbase;

            // 8 channels per lane = 2x b128 per corner
            v4f g00l = *(const v4f*)(xt + b00), g00h = *(const v4f*)(xt + b00 + 4);
            v4f g01l = *(const v4f*)(xt + b01), g01h = *(const v4f*)(xt + b01 + 4);
            v4f g10l = *(const v4f*)(xt + b10), g10h = *(const v4f*)(xt + b10 + 4);
            v4f g11l = *(const v4f*)(xt + b11), g11h = *(const v4f*)(xt + b11 + 4);

            v4f slo = g00l * wt00 + g01l * wt01 + g10l * wt10 + g11l * wt11;
            v4f shi = g00h * wt00 + g01h * wt01 + g10h * wt10 + g11h * wt11;

            v2f b0 = { slo.x, slo.y };
            v2f b1 = { slo.z, slo.w };
            v2f b2 = { shi.x, shi.y };
            v2f b3 = { shi.z, shi.w };
            acc = __builtin_amdgcn_wmma_f32_16x16x4_f32(false, afrag[k][0], false, b0,
                                                        (short)0, acc, false, false);
            acc = __builtin_amdgcn_wmma_f32_16x16x4_f32(false, afrag[k][1], false, b1,
                                                        (short)0, acc, false, false);
            acc = __builtin_amdgcn_wmma_f32_16x16x4_f32(false, afrag[k][2], false, b2,
                                                        (short)0, acc, false, false);
            acc = __builtin_amdgcn_wmma_f32_16x16x4_f32(false, afrag[k][3], false, b3,
                                                        (short)0, acc, false, false);
        }

        const int obase = h * (WD * LD) + w * LD + l0 + p;
        #pragma unroll
        for (int r = 0; r < 8; ++r)
            out[(size_t)(r + 8 * half) * SP + obase] = acc[r];
    }
}

// ---------------------------------------------------------------------------
// Fallback (no workspace): direct strided gather from x[C][H][W][L].
// ---------------------------------------------------------------------------
__global__ __launch_bounds__(256)
void dsconv_wmma_direct(const float* __restrict__ x,
                        const float* __restrict__ off,
                        const float* __restrict__ wgt,
                        float* __restrict__ out) {
    const int lane  = threadIdx.x & 31;
    const int half  = lane >> 4;
    const int p     = lane & 15;
    const int wpb   = blockDim.x >> 5;
    const int gwave = blockIdx.x * wpb + (threadIdx.x >> 5);
    const int nwav  = gridDim.x * wpb;

    v2f afrag[KT][4];
    {
        const int o = p;
        #pragma unroll
        for (int k = 0; k < KT; ++k)
            #pragma unroll
            for (int q = 0; q < 4; ++q) {
                const int c0 = 4 * q + 2 * half;
                afrag[k][q].x = wgt[(o * CC + c0    ) * KT + k];
                afrag[k][q].y = wgt[(o * CC + c0 + 1) * KT + k];
            }
    }

    for (int t = gwave; t < NT; t += nwav) {
        const int l0 = (t % TL) * 16;
        const int w  = (t / TL) % WD;
        const int h  =  t / (TL * WD);
        const int l  = l0 + p;
        const int sbase = h * (WD * LD) + w * LD + l;
        const int huni  = __builtin_amdgcn_readfirstlane(h);

        v8f acc = {};
        #pragma unroll
        for (int k = 0; k < KT; ++k) {
            const int hk = huni + k - PADH;
            if (hk < 0 || hk >= HD) continue;
            const float ow = off[(2 * k + 0) * SP + sbase];
            const float ol = off[(2 * k + 1) * SP + sbase];
            const float pw = (float)w + ow;
            const float pl = (float)l + ol;
            const float fw = floorf(pw), fl = floorf(pl);
            const float aw = pw - fw,   al = pl - fl;
            const int iw0 = (int)fw, iw1 = iw0 + 1;
            const int il0 = (int)fl, il1 = il0 + 1;
            const bool vw0 = (iw0 >= 0) & (iw0 < WD);
            const bool vw1 = (iw1 >= 0) & (iw1 < WD);
            const bool vl0 = (il0 >= 0) & (il0 < LD);
            const bool vl1 = (il1 >= 0) & (il1 < LD);
            float wt00 = (1.f - aw) * (1.f - al); if (!(vw0 & vl0)) wt00 = 0.f;
            float wt01 = (1.f - aw) * al;         if (!(vw0 & vl1)) wt01 = 0.f;
            float wt10 = aw * (1.f - al);         if (!(vw1 & vl0)) wt10 = 0.f;
            float wt11 = aw * al;                 if (!(vw1 & vl1)) wt11 = 0.f;
            const int cw0 = min(max(iw0, 0), WD - 1);
            const int cw1 = min(max(iw1, 0), WD - 1);
            const int cl0 = min(max(il0, 0), LD - 1);
            const int cl1 = min(max(il1, 0), LD - 1);
            const int hb  = hk * (WD * LD);
            const int b00 = hb + cw0 * LD + cl0;
            const int b01 = hb + cw0 * LD + cl1;
            const int b10 = hb + cw1 * LD + cl0;
            const int b11 = hb + cw1 * LD + cl1;

            #pragma unroll
            for (int q = 0; q < 4; ++q) {
                const int c0 = 4 * q + 2 * half;
                const float* xc0 = x + (size_t)c0 * SP;
                const float* xc1 = x + (size_t)(c0 + 1) * SP;
                v2f b;
                b.x = wt00 * xc0[b00] + wt01 * xc0[b01] + wt10 * xc0[b10] + wt11 * xc0[b11];
                b.y = wt00 * xc1[b00] + wt01 * xc1[b01] + wt10 * xc1[b10] + wt11 * xc1[b11];
                acc = __builtin_amdgcn_wmma_f32_16x16x4_f32(false, afrag[k][q], false, b,
                                                            (short)0, acc, false, false);
            }
        }

        const int obase = h * (WD * LD) + w * LD + l0 + p;
        #pragma unroll
        for (int r = 0; r < 8; ++r)
            out[(size_t)(r + 8 * half) * SP + obase] = acc[r];
    }
}

extern "C" void kernel_launch(void* const* d_in, const int* in_sizes, int n_in,
                              void* d_out, int out_size, void* d_ws, size_t ws_size,
                              hipStream_t stream) {
    const float* x   = (const float*)d_in[0];   // [1,16,80,80,80]
    const float* off = (const float*)d_in[1];   // [1,18,80,80,80]
    const float* wgt = (const float*)d_in[2];   // [16,16,9,1,1]
    float* out = (float*)d_out;                 // [1,16,80,80,80]

    const size_t need = (size_t)SP * CC * sizeof(float);   // 32.768 MB
    if (d_ws != nullptr && ws_size >= need) {
        float* xt = (float*)d_ws;
        pack_xt<<<(SP + 255) / 256, 256, 0, stream>>>(x, xt);
        dsconv_wmma_packed<<<1000, 256, 0, stream>>>(xt, off, wgt, out);
    } else {
        dsconv_wmma_direct<<<1000, 256, 0, stream>>>(x, off, wgt, out);
    }
    (void)in_sizes; (void)n_in; (void)out_size;
}